// HebbianLayer_57681410786163
// MI455X (gfx1250) — compile-verified
//
#include <hip/hip_runtime.h>

#define LR 0.001f
#define BATCH 256
#define IN 1024
#define OUT 512

#define NWG 16               // cooperating workgroups in the scan
#define COLS 64              // IN / NWG columns of W per workgroup
#define TPB 64               // one thread per owned column
#define RPT 8                // OUT / TPB rows per thread in phase 1
#define SLAB_STRIDE (COLS + 1)   // LDS pad -> conflict-free in both phases

typedef __attribute__((ext_vector_type(2))) float v2f;
typedef __attribute__((ext_vector_type(8))) float v8f;

// ---------------------------------------------------------------------------
// Forward: y[256,512] = x[256,1024] @ W^T + bias, via V_WMMA_F32_16X16X4_F32.
// One wave per 16x16 output tile. ISA 32-bit A(16x4)/B(4x16) fragment layout:
// lane L holds elements (k + 2*(L>=16), k + 1 + 2*(L>=16)) of row L%16.
// ---------------------------------------------------------------------------
__global__ __launch_bounds__(32) void fwd_wmma(const float* __restrict__ x,
                                               const float* __restrict__ w,
                                               const float* __restrict__ bias,
                                               float* __restrict__ y) {
  const int lane = threadIdx.x;
  const int lm = lane & 15;
  const int hi = lane >> 4;                 // 0 or 1
  const int n0 = blockIdx.x * 16;           // output-feature tile
  const int m0 = blockIdx.y * 16;           // batch tile

  const float* xrow = x + (size_t)(m0 + lm) * IN + 2 * hi;  // A row (x)
  const float* wrow = w + (size_t)(n0 + lm) * IN + 2 * hi;  // B row (W row n)

  v8f c = {0.f, 0.f, 0.f, 0.f, 0.f, 0.f, 0.f, 0.f};
  for (int k = 0; k < IN; k += 4) {
    v2f a = *(const v2f*)(xrow + k);
    v2f b = *(const v2f*)(wrow + k);
    c = __builtin_amdgcn_wmma_f32_16x16x4_f32(false, a, false, b,
                                              (short)0, c, false, false);
  }
  // C/D layout: VGPR r = row (r + 8*hi), column lm.
  const float bv = bias[n0 + lm];
#pragma unroll
  for (int r = 0; r < 8; ++r)
    y[(size_t)(m0 + r + 8 * hi) * OUT + n0 + lm] = c[r] + bv;
}

// ---------------------------------------------------------------------------
// Device-scope sense-reversing barrier across NWG workgroups.
// ---------------------------------------------------------------------------
__device__ __forceinline__ void grid_barrier(unsigned* bar) {
  __syncthreads();
  if (threadIdx.x == 0) {
    __threadfence();
    unsigned gen = __hip_atomic_load(bar + 1, __ATOMIC_RELAXED,
                                     __HIP_MEMORY_SCOPE_AGENT);
    unsigned arrived = __hip_atomic_fetch_add(bar, 1u, __ATOMIC_ACQ_REL,
                                              __HIP_MEMORY_SCOPE_AGENT) + 1u;
    if (arrived == NWG) {
      __hip_atomic_store(bar, 0u, __ATOMIC_RELAXED, __HIP_MEMORY_SCOPE_AGENT);
      __hip_atomic_store(bar + 1, gen + 1u, __ATOMIC_RELEASE,
                         __HIP_MEMORY_SCOPE_AGENT);
    } else {
      while (__hip_atomic_load(bar + 1, __ATOMIC_ACQUIRE,
                               __HIP_MEMORY_SCOPE_AGENT) == gen) {
        __builtin_amdgcn_s_sleep(2);
      }
    }
    __threadfence();
  }
  __syncthreads();
}

// ---------------------------------------------------------------------------
// Sequential Sanger scan. WG g owns columns [g*64, g*64+64) of W, resident in
// LDS for all 256 samples. Per sample: partial GEMV -> barrier -> yi reduce ->
// private per-column prefix-scan update (exclusive scan form of tril@W).
// ---------------------------------------------------------------------------
__global__ __launch_bounds__(TPB) void sanger_scan(
    const float* __restrict__ x, const float* __restrict__ w0,
    const float* __restrict__ bias, float* __restrict__ wOut,
    float* __restrict__ partial, unsigned* __restrict__ bar) {
  extern __shared__ float Wslab[];          // [OUT][SLAB_STRIDE] = 130 KB
  __shared__ float yis[OUT];
  __shared__ float xs[COLS];
  __shared__ float bls[OUT];

  const int tid = threadIdx.x;
  const int g = blockIdx.x;
  const int myCol = g * COLS + tid;

  // Load our 64-column slab of W (coalesced 256B per row) and bias into LDS.
  for (int j = 0; j < OUT; ++j)
    Wslab[j * SLAB_STRIDE + tid] = w0[(size_t)j * IN + myCol];
#pragma unroll
  for (int r = 0; r < RPT; ++r)
    bls[tid + r * TPB] = bias[tid + r * TPB];
  __syncthreads();

  for (int t = 0; t < BATCH; ++t) {
    // Stage this sample's x chunk; prefetch next sample's chunk.
    xs[tid] = x[(size_t)t * IN + myCol];
    if (t + 1 < BATCH)
      __builtin_prefetch(x + (size_t)(t + 1) * IN + myCol, 0, 3);
    __syncthreads();

    // Phase 1: partial[j] = sum over our 64 columns of W[j,i]*x[t,i].
    float acc[RPT];
#pragma unroll
    for (int r = 0; r < RPT; ++r) acc[r] = 0.f;
    for (int ii = 0; ii < COLS; ++ii) {
      const float xv = xs[ii];                       // LDS broadcast
#pragma unroll
      for (int r = 0; r < RPT; ++r)                  // conflict-free (pad)
        acc[r] += Wslab[(tid + r * TPB) * SLAB_STRIDE + ii] * xv;
    }
    float* pb = partial + (size_t)(t & 1) * (NWG * OUT) + (size_t)g * OUT;
#pragma unroll
    for (int r = 0; r < RPT; ++r)
      pb[tid + r * TPB] = acc[r];

    // One barrier per sample (partials are double-buffered on t&1).
    grid_barrier(bar);

    // Reduce 16 partials + bias -> yi, into LDS.
    const float* pall = partial + (size_t)(t & 1) * (NWG * OUT);
#pragma unroll
    for (int r = 0; r < RPT; ++r) {
      const int j = tid + r * TPB;
      float s = bls[j];
      for (int g2 = 0; g2 < NWG; ++g2)
        s += pall[(size_t)g2 * OUT + j];
      yis[j] = s;
    }
    __syncthreads();

    // Phase 2: private sequential column recurrence (no sync needed inside):
    //   W[j,i] += LR*yi[j]*(x[i] - c);  c += yi[j]*W_old[j,i]
    float c = 0.f;
    const float xv = xs[tid];
    for (int j = 0; j < OUT; ++j) {
      const float yj = yis[j];
      const float wold = Wslab[j * SLAB_STRIDE + tid];
      Wslab[j * SLAB_STRIDE + tid] = fmaf(LR * yj, xv - c, wold);
      c = fmaf(yj, wold, c);
    }
    __syncthreads();  // slab/xs/yis stable before next sample
  }

  // Write the final weights (inputs are never mutated).
  for (int j = 0; j < OUT; ++j)
    wOut[(size_t)j * IN + myCol] = Wslab[j * SLAB_STRIDE + tid];
}

__global__ void init_bar(unsigned* bar) {
  bar[0] = 0u;
  bar[1] = 0u;
}

extern "C" void kernel_launch(void* const* d_in, const int* in_sizes, int n_in,
                              void* d_out, int out_size, void* d_ws,
                              size_t ws_size, hipStream_t stream) {
  const float* x = (const float*)d_in[0];      // [256,1024]
  const float* w = (const float*)d_in[1];      // [512,1024]
  const float* bias = (const float*)d_in[2];   // [512]
  float* y = (float*)d_out;                    // [256,512]
  float* wOut = (float*)d_out + (size_t)BATCH * OUT;  // [512,1024]

  unsigned* bar = (unsigned*)d_ws;                        // 2 uints
  float* partial = (float*)((char*)d_ws + 256);           // 2*16*512 floats

  init_bar<<<1, 1, 0, stream>>>(bar);

  dim3 fgrid(OUT / 16, BATCH / 16);
  fwd_wmma<<<fgrid, 32, 0, stream>>>(x, w, bias, y);

  const size_t slabBytes = (size_t)OUT * SLAB_STRIDE * sizeof(float);
  sanger_scan<<<NWG, TPB, slabBytes, stream>>>(x, w, bias, wOut, partial, bar);
}